// MulitmodalCompactBilinearPool_37374805410017
// MI455X (gfx1250) — compile-verified
//
#include <hip/hip_runtime.h>
#include <math.h>

// ---------------------------------------------------------------------------
// MultimodalCompactBilinearPool on MI455X (gfx1250, wave32, WMMA)
//
//   y0 = countsketch(x0,C0); y1 = countsketch(x1,C1)        (sparse scatter)
//   G0 = DFT16000(y0); G1 = DFT16000(y1)                    (four-step, WMMA)
//   P  = conj(Re0*Re1 + j Im0*Im1)/16000                    (elementwise)
//   phi= Re(DFT16000(P))                                    (same engine)
//
// DFT16000 = 125-point stage (fused complex GEMM + twiddle epilogue) ->
//            128-point stage (fused complex GEMM). Each fused stage is ONE
//            kernel: both DFT bases (cos, -sin) in 128 KB LDS, Re and Im
//            accumulated together with V_WMMA_F32_16X16X4_F32, A plane read
//            once. 16 waves/block x (16 rows x 64 cols) keeps the Re+Im
//            accumulator set at 64 VGPRs -> no scratch spills (round-2 issue).
// ---------------------------------------------------------------------------

typedef float v2f __attribute__((ext_vector_type(2)));
typedef float v8f __attribute__((ext_vector_type(8)));

#define ORIG_DIM 2048
#define PROJ     16000
#define PPAD     16384                       // padded length (128*128)
#define BATCH    512
#define PLANEF   ((size_t)BATCH * PPAD)      // floats per [512][16384] plane
#define TWO_PI   6.28318530717958647692f

// ------------------------------- tables ------------------------------------
__global__ void k_tables(float* __restrict__ c125, float* __restrict__ s125,
                         float* __restrict__ c128, float* __restrict__ s128) {
  int idx = blockIdx.x * blockDim.x + threadIdx.x;   // 0..16383
  if (idx >= PPAD) return;
  int r = idx >> 7, c = idx & 127;
  float vc = 0.f, vs = 0.f;
  if (r < 125 && c < 125) {
    int m = (r * c) % 125;
    float ang = TWO_PI * (float)m / 125.f;
    vc = cosf(ang); vs = -sinf(ang);
  }
  c125[idx] = vc; s125[idx] = vs;
  int m8 = (r * c) & 127;
  float a8 = TWO_PI * (float)m8 / 128.f;
  c128[idx] = cosf(a8); s128[idx] = -sinf(a8);
}

// --------------------- count-sketch structure extraction -------------------
__global__ void k_extract(const float* __restrict__ C0,
                          const float* __restrict__ C1,
                          int* __restrict__ col0, float* __restrict__ sgn0,
                          int* __restrict__ col1, float* __restrict__ sgn1) {
  int row = blockIdx.x;
  const float* C = (blockIdx.y == 0) ? C0 : C1;
  int*   col = (blockIdx.y == 0) ? col0 : col1;
  float* sgn = (blockIdx.y == 0) ? sgn0 : sgn1;
  int lane = threadIdx.x;
  int enc = -1;
  for (int i = lane; i < PROJ; i += 32) {
    float v = C[(size_t)row * PROJ + i];
    if (v != 0.f) enc = (i << 1) | (v > 0.f ? 1 : 0);
  }
  for (int off = 16; off; off >>= 1) {
    int o = __shfl_xor(enc, off, 32);
    enc = (o > enc) ? o : enc;
  }
  if (lane == 0) {
    if (enc < 0) { col[row] = 0; sgn[row] = 0.f; }
    else         { col[row] = enc >> 1; sgn[row] = (enc & 1) ? 1.f : -1.f; }
  }
}

// ----------------------- deterministic sketch scatter ----------------------
__global__ void __launch_bounds__(256)
k_scatter(const float* __restrict__ x0, const float* __restrict__ x1,
          const int* __restrict__ col0, const float* __restrict__ sgn0,
          const int* __restrict__ col1, const float* __restrict__ sgn1,
          float* __restrict__ y0, float* __restrict__ y1) {
  __shared__ float ylds[PPAD];                       // 64 KB
  int b = blockIdx.x, t = threadIdx.x;
  const float* x   = (blockIdx.y == 0) ? x0 : x1;
  const int*   col = (blockIdx.y == 0) ? col0 : col1;
  const float* sgn = (blockIdx.y == 0) ? sgn0 : sgn1;
  float*       y   = (blockIdx.y == 0) ? y0 : y1;
  for (int i = t; i < PPAD; i += 256) ylds[i] = 0.f;
  __syncthreads();
  int lo = t << 6;
  for (int d = 0; d < ORIG_DIM; ++d) {
    int   c = col[d];                                // uniform -> scalar load
    float v = sgn[d] * x[(size_t)b * ORIG_DIM + d];
    if ((unsigned)(c - lo) < 64u) ylds[c] += v;
  }
  __syncthreads();
  for (int i = t; i < PPAD; i += 256) y[(size_t)b * PPAD + i] = ylds[i];
}

// ------------------------- fused complex WMMA DFT stage --------------------
// D = A_cplx x (C + iS) per 128x128x128 batch plane, f32 WMMA 16x16x4.
//   Dre = Ar*C - Ai*S ; Dim = Ar*S + Ai*C   (Ai absent if !HAS_IM)
// A planes are [k*128+m]-contiguous (true for both four-step stages).
// 512 threads = 16 waves: wave w -> rows [(w>>1)*16, +16), cols [(w&1)*64,+64)
// => 4 N-tiles, Re+Im acc = 8 x v8f = 64 VGPRs (no spills).
// TW: epilogue rotate by W_16000^{row*col} fused into the store loop so each
// accumulator dies immediately (row*col <= 16129 -> one conditional subtract).
// f32 WMMA has no A/B negation (NEG=CNeg only) -> negate Im A-frag in VALU.
// ROWS_VALID as template arg: guard disappears for 128, uniform fast path
// for 125.
template <bool HAS_IM, bool OUT_IM, bool TW, int ROWS_VALID>
__global__ void __launch_bounds__(512)
k_dft(const float* __restrict__ Ar, const float* __restrict__ Ai,
      const float* __restrict__ Cmat, const float* __restrict__ Smat,
      float* __restrict__ Dre, float* __restrict__ Dim,
      int dRowStride, int dColStride, size_t dPlane) {
  __shared__ float Bc[128 * 128];                    // 64 KB (cos basis)
  __shared__ float Bn[128 * 128];                    // 64 KB (-sin basis)
  const int tid  = threadIdx.x;
  const int lane = tid & 31;
  const int wave = tid >> 5;                         // 0..15
  const int half = lane >> 4;                        // 0: lanes 0-15, 1: 16-31
  const int l16  = lane & 15;
  const size_t pa = (size_t)blockIdx.x * PPAD;
  const float* Agr = Ar + pa;
  const float* Agi = HAS_IM ? (Ai + pa) : nullptr;
  float* Dr = Dre + (size_t)blockIdx.x * dPlane;
  float* Di = OUT_IM ? (Dim + (size_t)blockIdx.x * dPlane) : nullptr;

  for (int i = tid; i < 128 * 128; i += 512) {
    Bc[i] = Cmat[i];
    Bn[i] = Smat[i];
  }
  __syncthreads();

  const int m0 = (wave >> 1) * 16;                   // row block
  const int n0 = (wave & 1) * 64;                    // col half
  const int mr = m0 + l16;

  v8f accRe[4], accIm[4];
  for (int i = 0; i < 4; ++i)
    for (int j = 0; j < 8; ++j) { accRe[i][j] = 0.f; accIm[i][j] = 0.f; }

  // A-frag (ISA 16x4 f32): VGPR0 = K=kk, VGPR1 = K=kk+1, kk = k0+2*(lane>=16),
  // rows M = lane&15. B-frag mirrored on the N dimension.
  for (int k0 = 0; k0 < 128; k0 += 4) {
    const int kk = k0 + half * 2;
    v2f arf;
    arf.x = Agr[kk * 128 + mr];
    arf.y = Agr[(kk + 1) * 128 + mr];
    v2f aif, ainf;
    if (HAS_IM) {
      aif.x = Agi[kk * 128 + mr];
      aif.y = Agi[(kk + 1) * 128 + mr];
      ainf.x = -aif.x;
      ainf.y = -aif.y;
    }
#pragma unroll
    for (int i = 0; i < 4; ++i) {
      const int nn = n0 + i * 16 + l16;
      v2f bc, bs;
      bc.x = Bc[kk * 128 + nn];
      bc.y = Bc[(kk + 1) * 128 + nn];
      bs.x = Bn[kk * 128 + nn];
      bs.y = Bn[(kk + 1) * 128 + nn];
      accRe[i] = __builtin_amdgcn_wmma_f32_16x16x4_f32(
          false, arf, false, bc, (short)0, accRe[i], false, false);
      if (HAS_IM)
        accRe[i] = __builtin_amdgcn_wmma_f32_16x16x4_f32(
            false, ainf, false, bs, (short)0, accRe[i], false, false);
      if (OUT_IM) {
        accIm[i] = __builtin_amdgcn_wmma_f32_16x16x4_f32(
            false, arf, false, bs, (short)0, accIm[i], false, false);
        if (HAS_IM)
          accIm[i] = __builtin_amdgcn_wmma_f32_16x16x4_f32(
              false, aif, false, bc, (short)0, accIm[i], false, false);
      }
    }
  }

  // ---- epilogue: (optional) twiddle rotate fused with stores, per tile ----
  const bool fullWave = (ROWS_VALID == 128) || (m0 + 16 <= ROWS_VALID);
  for (int i = 0; i < 4; ++i) {
    const int colN = n0 + i * 16 + l16;
    if (fullWave) {
#pragma unroll
      for (int j = 0; j < 8; ++j) {
        int row = m0 + j + half * 8;
        float re = accRe[i][j];
        float im = OUT_IM ? accIm[i][j] : 0.f;
        if (TW) {
          int t = row * colN;
          t = (t >= 16000) ? t - 16000 : t;          // t <= 16129
          float a = (TWO_PI / 16000.f) * (float)t;
          float tc = cosf(a), ts = -sinf(a);
          float r2 = re * tc - im * ts;
          im = re * ts + im * tc;
          re = r2;
        }
        size_t ad = (size_t)row * dRowStride + (size_t)colN * dColStride;
        Dr[ad] = re;
        if (OUT_IM) Di[ad] = im;
      }
    } else {
#pragma unroll
      for (int j = 0; j < 8; ++j) {
        int row = m0 + j + half * 8;
        if (row < ROWS_VALID) {
          float re = accRe[i][j];
          float im = OUT_IM ? accIm[i][j] : 0.f;
          if (TW) {
            int t = row * colN;
            t = (t >= 16000) ? t - 16000 : t;
            float a = (TWO_PI / 16000.f) * (float)t;
            float tc = cosf(a), ts = -sinf(a);
            float r2 = re * tc - im * ts;
            im = re * ts + im * tc;
            re = r2;
          }
          size_t ad = (size_t)row * dRowStride + (size_t)colN * dColStride;
          Dr[ad] = re;
          if (OUT_IM) Di[ad] = im;
        }
      }
    }
  }
}

// --------------------- spectral product (conjugated, scaled) ---------------
__global__ void k_product(float* __restrict__ g0re, float* __restrict__ g0im,
                          const float* __restrict__ g1re,
                          const float* __restrict__ g1im) {
  size_t idx = (size_t)blockIdx.x * blockDim.x + threadIdx.x;
  int j = (int)(idx & (PPAD - 1));
  const float scale = 1.f / 16000.f;
  if (j < PROJ) {
    g0re[idx] = g0re[idx] * g1re[idx] * scale;
    g0im[idx] = -(g0im[idx] * g1im[idx]) * scale;
  } else {
    g0re[idx] = 0.f;
    g0im[idx] = 0.f;
  }
}

// ---------------------------------------------------------------------------
extern "C" void kernel_launch(void* const* d_in, const int* in_sizes, int n_in,
                              void* d_out, int out_size, void* d_ws,
                              size_t ws_size, hipStream_t stream) {
  const float* x0 = (const float*)d_in[0];
  const float* x1 = (const float*)d_in[1];
  const float* C0 = (const float*)d_in[2];
  const float* C1 = (const float*)d_in[3];
  float* out = (float*)d_out;

  float* ws = (float*)d_ws;
  float* c125 = ws;
  float* s125 = ws + 1 * PPAD;
  float* c128 = ws + 2 * PPAD;
  float* s128 = ws + 3 * PPAD;
  int*   col0 = (int*)(ws + 4 * PPAD);
  float* sgn0 = ws + 4 * PPAD + 2048;
  int*   col1 = (int*)(ws + 4 * PPAD + 4096);
  float* sgn1 = ws + 4 * PPAD + 6144;
  float* base = ws + 8 * PPAD;                   // plane area (6 x 32 MB)
  float* P2 = base + 0 * PLANEF;                 // stage-A re
  float* P3 = base + 1 * PLANEF;                 // stage-A im
  float* P4 = base + 2 * PLANEF;                 // y0 -> G0re -> Pre
  float* P5 = base + 3 * PLANEF;                 // G0im -> Pim
  float* P6 = base + 4 * PLANEF;                 // y1 -> G1re
  float* P7 = base + 5 * PLANEF;                 // G1im

  const int TB = 256;
  const int gElem = (int)((512 * (size_t)PPAD) / TB);    // 32768

  k_tables<<<PPAD / TB, TB, 0, stream>>>(c125, s125, c128, s128);
  k_extract<<<dim3(ORIG_DIM, 2), 32, 0, stream>>>(C0, C1, col0, sgn0, col1, sgn1);
  k_scatter<<<dim3(BATCH, 2), TB, 0, stream>>>(x0, x1, col0, sgn0, col1, sgn1, P4, P6);

  // ---- forward DFT of y0: stage A (real->cplx, twiddled), stage C ----
  k_dft<false, true, true, 128><<<BATCH, 512, 0, stream>>>(
      P4, nullptr, c125, s125, P2, P3, 128, 1, (size_t)PPAD);
  k_dft<true, true, false, 125><<<BATCH, 512, 0, stream>>>(
      P2, P3, c128, s128, P4, P5, 1, 125, (size_t)PPAD);           // G0

  // ---- forward DFT of y1 ----
  k_dft<false, true, true, 128><<<BATCH, 512, 0, stream>>>(
      P6, nullptr, c125, s125, P2, P3, 128, 1, (size_t)PPAD);
  k_dft<true, true, false, 125><<<BATCH, 512, 0, stream>>>(
      P2, P3, c128, s128, P6, P7, 1, 125, (size_t)PPAD);           // G1

  // ---- Pc = conj(Re0*Re1 + j Im0*Im1)/N, in-place into P4/P5 ----
  k_product<<<gElem, TB, 0, stream>>>(P4, P5, P6, P7);

  // ---- inverse: forward-DFT of Pc, real part only, straight to d_out ----
  k_dft<true, true, true, 128><<<BATCH, 512, 0, stream>>>(
      P4, P5, c125, s125, P2, P3, 128, 1, (size_t)PPAD);
  k_dft<true, false, false, 125><<<BATCH, 512, 0, stream>>>(
      P2, P3, c128, s128, out, nullptr, 1, 125, (size_t)PROJ);
}